// S4CondBlock_1855425872424
// MI455X (gfx1250) — compile-verified
//
#include <hip/hip_runtime.h>
#include <hip/hip_bf16.h>
#include <math.h>

// S4/DSSM block for MI455X (gfx1250, wave32).
//   B=8, H=64, L=32768, N=16 modes.
// Phase 1: u = tanh(W x + b)   -> fp32 WMMA (v_wmma_f32_16x16x4_f32) GEMM.
// Phase 2: y = causal conv(u, K) done as 16-mode complex linear recurrence
//          (chunked parallel scan), fused with D-skip, tanh, residual.
// Bandwidth-bound (~470 MB of traffic -> ~20 us at 23.3 TB/s); compute
// (~6.5 GFLOP) is negligible on this chip.

typedef __attribute__((ext_vector_type(2))) float v2f;
typedef __attribute__((ext_vector_type(8))) float v8f;

#define Bsz 8
#define Hsz 64
#define Lsz 32768
#define Nmode 16
#define NTHREADS 256
#define CHUNK (Lsz / NTHREADS)   // 128

// ---------------------------------------------------------------------------
// Kernel 1: channel mix  u[b,g,l] = tanh( sum_h W[g,h] * x[b,h,l] + bias[g] )
// One wave computes M=64 (4 tiles of 16) x N=16 columns; K=64 in 16 steps of 4.
// A (16x4 f32) layout per ISA: lanes 0-15 M=0..15 (both halves), VGPR r holds
// K = r + 2*half -> contiguous {K,K+1} pair per lane == one float2 load of W.
// B (4x16 f32): VGPR r holds row K = r + 2*half, col N = lane%16 (coalesced x).
// C/D (16x16 f32): VGPR r -> row M = r + 8*half, col N = lane%16.
// ---------------------------------------------------------------------------
__global__ __launch_bounds__(NTHREADS)
void k_channel_mix(const float* __restrict__ x, const float* __restrict__ W,
                   const float* __restrict__ bias, float* __restrict__ u)
{
    const int tid  = threadIdx.x;
    const int wave = tid >> 5;
    const int lane = tid & 31;
    const int half = lane >> 4;
    const int lp   = lane & 15;

    const int b  = blockIdx.y;
    const int l0 = blockIdx.x * 128 + wave * 16;
    const float* xb = x + (size_t)b * Hsz * Lsz;

    v8f acc[4] = {};

    for (int k0 = 0; k0 < Hsz; k0 += 4) {
        // B operand: rows k0..k0+3 of x, columns l0..l0+15 (coalesced per half)
        v2f bb;
        bb.x = xb[(size_t)(k0 + 2 * half + 0) * Lsz + l0 + lp];
        bb.y = xb[(size_t)(k0 + 2 * half + 1) * Lsz + l0 + lp];
#pragma unroll
        for (int mt = 0; mt < 4; ++mt) {
            // A operand: W[(mt*16+lp), k0 + 2*half + {0,1}] -- 8B-aligned pair
            const float* wrow = W + (size_t)(mt * 16 + lp) * Hsz + k0 + 2 * half;
            v2f aa = *(const v2f*)wrow;
            acc[mt] = __builtin_amdgcn_wmma_f32_16x16x4_f32(
                false, aa, false, bb, (short)0, acc[mt], false, false);
        }
    }

#pragma unroll
    for (int mt = 0; mt < 4; ++mt) {
#pragma unroll
        for (int r = 0; r < 8; ++r) {
            const int g = mt * 16 + r + half * 8;
            const float v = acc[mt][r] + bias[g];
            u[((size_t)b * Hsz + g) * Lsz + l0 + lp] = tanhf(v);
        }
    }
}

// ---------------------------------------------------------------------------
// Kernel 2: per-(b,h) 16-mode complex scan (blockwise chunked), fused epilogue.
//   s_n[l] = z_n * s_n[l-1] + u[l],   y[l] = 2*Re( sum_n w_n * s_n[l] )
//   out    = tanh(y + u*D[h]) + x*res_w[h]
// One 256-thread block per (b,h); each thread owns a 128-element chunk.
// ---------------------------------------------------------------------------
__global__ __launch_bounds__(NTHREADS)
void k_dssm_scan(const float* __restrict__ u, const float* __restrict__ x,
                 const float* __restrict__ log_dt,
                 const float* __restrict__ C_re, const float* __restrict__ C_im,
                 const float* __restrict__ log_A_real,
                 const float* __restrict__ A_imag,
                 const float* __restrict__ Dp, const float* __restrict__ res_w,
                 float* __restrict__ out)
{
    __shared__ float zre[Nmode], zim[Nmode];       // z = exp(dt*A)
    __shared__ float wre[Nmode], wim[Nmode];       // Cmod = C*(z-1)/A
    __shared__ float Zre[Nmode], Zim[Nmode];       // Z = z^CHUNK
    __shared__ float sAre[NTHREADS][Nmode + 1];    // +1 pad: kill bank conflicts
    __shared__ float sAim[NTHREADS][Nmode + 1];

    const int tid = threadIdx.x;
    const int bh  = blockIdx.x;          // 0 .. B*H-1
    const int h   = bh & (Hsz - 1);

    if (tid < Nmode) {
        const int n = tid;
        const float dt  = expf(log_dt[h]);
        const float Are = -expf(log_A_real[h * Nmode + n]);
        const float Aim = A_imag[h * Nmode + n];
        const float dre = Are * dt, dim = Aim * dt;          // dt*A
        const float er  = expf(dre);
        const float zr  = er * cosf(dim), zi = er * sinf(dim);
        zre[n] = zr; zim[n] = zi;
        // Cmod = C * (z - 1) / A   (complex divide via conj(A)/|A|^2)
        const float e1r = zr - 1.0f, e1i = zi;
        const float cr = C_re[h * Nmode + n], ci = C_im[h * Nmode + n];
        const float nr = cr * e1r - ci * e1i;
        const float ni = cr * e1i + ci * e1r;
        const float den = Are * Are + Aim * Aim;
        wre[n] = (nr * Are + ni * Aim) / den;
        wim[n] = (ni * Are - nr * Aim) / den;
        // Z = z^CHUNK = exp(dt*A*CHUNK)
        const float er2 = expf(dre * (float)CHUNK);
        const float ph  = dim * (float)CHUNK;
        Zre[n] = er2 * cosf(ph);
        Zim[n] = er2 * sinf(ph);
    }
    __syncthreads();

    float lzr[Nmode], lzi[Nmode];
#pragma unroll
    for (int n = 0; n < Nmode; ++n) { lzr[n] = zre[n]; lzi[n] = zim[n]; }

    const size_t base = (size_t)bh * Lsz + (size_t)tid * CHUNK;

    // ---- pass 1: local chunk scan, record end state -----------------------
    float sr[Nmode] = {}, si[Nmode] = {};
    for (int j = 0; j < CHUNK; ++j) {
        const float uv = u[base + j];
#pragma unroll
        for (int n = 0; n < Nmode; ++n) {
            const float tr = lzr[n] * sr[n] - lzi[n] * si[n] + uv;
            const float ti = lzr[n] * si[n] + lzi[n] * sr[n];
            sr[n] = tr; si[n] = ti;
        }
    }
#pragma unroll
    for (int n = 0; n < Nmode; ++n) { sAre[tid][n] = sr[n]; sAim[tid][n] = si[n]; }
    __syncthreads();

    // ---- cross-chunk exclusive scan (16 lanes, one mode each) -------------
    if (tid < Nmode) {
        const int n = tid;
        const float Zr = Zre[n], Zi = Zim[n];
        float pr = 0.0f, pi = 0.0f;
        for (int c = 0; c < NTHREADS; ++c) {
            const float tr = sAre[c][n], ti = sAim[c][n];
            sAre[c][n] = pr; sAim[c][n] = pi;
            const float npr = Zr * pr - Zi * pi + tr;
            const float npi = Zr * pi + Zi * pr + ti;
            pr = npr; pi = npi;
        }
    }
    __syncthreads();

    // ---- pass 2: rescan with prefix seed + fused epilogue -----------------
    float lwr[Nmode], lwi[Nmode];
#pragma unroll
    for (int n = 0; n < Nmode; ++n) {
        sr[n] = sAre[tid][n]; si[n] = sAim[tid][n];
        lwr[n] = wre[n];      lwi[n] = wim[n];
    }
    const float Dh = Dp[h], rwh = res_w[h];
    for (int j = 0; j < CHUNK; ++j) {
        const float uv = u[base + j];
        float acc = 0.0f;
#pragma unroll
        for (int n = 0; n < Nmode; ++n) {
            const float tr = lzr[n] * sr[n] - lzi[n] * si[n] + uv;
            const float ti = lzr[n] * si[n] + lzi[n] * sr[n];
            sr[n] = tr; si[n] = ti;
            acc += lwr[n] * tr - lwi[n] * ti;
        }
        const float y = tanhf(2.0f * acc + uv * Dh);
        out[base + j] = y + x[base + j] * rwh;
    }
}

// ---------------------------------------------------------------------------
extern "C" void kernel_launch(void* const* d_in, const int* in_sizes, int n_in,
                              void* d_out, int out_size, void* d_ws, size_t ws_size,
                              hipStream_t stream)
{
    (void)in_sizes; (void)n_in; (void)out_size; (void)ws_size;
    const float* x      = (const float*)d_in[0];   // (B,H,L)
    const float* W      = (const float*)d_in[1];   // (H,H)
    const float* bias   = (const float*)d_in[2];   // (H)
    const float* log_dt = (const float*)d_in[3];   // (H)
    const float* C_re   = (const float*)d_in[4];   // (H,N)
    const float* C_im   = (const float*)d_in[5];   // (H,N)
    const float* lAr    = (const float*)d_in[6];   // (H,N)
    const float* Aim    = (const float*)d_in[7];   // (H,N)
    const float* Dp     = (const float*)d_in[8];   // (H)
    const float* rw     = (const float*)d_in[9];   // (H)
    float* out = (float*)d_out;                    // (B,H,L)
    float* u   = (float*)d_ws;                     // scratch: B*H*L floats (64 MB)

    dim3 g1(Lsz / 128, Bsz);
    k_channel_mix<<<g1, NTHREADS, 0, stream>>>(x, W, bias, u);
    k_dssm_scan<<<Bsz * Hsz, NTHREADS, 0, stream>>>(u, x, log_dt, C_re, C_im,
                                                    lAr, Aim, Dp, rw, out);
}